// DeformConv3d_14568529068119
// MI455X (gfx1250) — compile-verified
//
#include <hip/hip_runtime.h>
#include <hip/hip_bf16.h>

// ---------------------------------------------------------------------------
// DeformConv3d for MI455X (gfx1250, wave32).
// Matrix work done with V_WMMA_F32_16X16X4_F32 (fp32 in/out, K=4 per op).
// ---------------------------------------------------------------------------

typedef __attribute__((ext_vector_type(2))) float v2f;
typedef __attribute__((ext_vector_type(8))) float v8f;

#define BB     2
#define CC     32
#define DIM    24
#define VOX    (DIM*DIM*DIM)      // 13824
#define NPTS   27
#define OC1    81                 // 3*NPTS offset channels
#define KDIM   864                // CC*NPTS
#define OUTC   64
#define SPAD   26                 // padded spatial extent

static __device__ __forceinline__ int iclamp(int a, int lo, int hi) {
    return a < lo ? lo : (a > hi ? hi : a);
}
static __device__ __forceinline__ int imin(int a, int b) { return a < b ? a : b; }

// ---------------------------------------------------------------------------
// Kernel 1: offset-predicting 3x3x3 conv as implicit GEMM with fp32 WMMA.
// grid = (576, 2) : blockIdx.x = x0*24 + x1, blockIdx.y = b. block = 384 (12 waves).
// Each wave owns one 16x16 tile: mt = wave/2 in [0,6), nt = wave&1 in {0,1}.
// Per-row (c,k0,k1,k2) decomposition is precomputed into an LDS table so the
// inner loop is just: table b64 load + 2 slab b32 loads + A b64 load + WMMA.
// Writes offs[b][o][v] (o<81) to workspace.
// ---------------------------------------------------------------------------
__global__ __launch_bounds__(384)
void k1_offset_conv(const float* __restrict__ x, const float* __restrict__ pw,
                    const float* __restrict__ pb, float* __restrict__ offs)
{
    __shared__ float    slab[CC * 3 * 3 * SPAD];   // 7488 floats = 29.25 KB
    __shared__ unsigned tab[KDIM];                 // 3456 B

    const int b  = blockIdx.y;
    const int x0 = blockIdx.x / DIM;
    const int x1 = blockIdx.x % DIM;
    const float* xb = x + (size_t)b * CC * VOX;

    // Row -> slab-base table (base already includes k2; add column later).
    for (int r = threadIdx.x; r < KDIM; r += 384) {
        int c = r / NPTS, t = r % NPTS;
        tab[r] = (unsigned)(((c * 3 + t / 9) * 3 + (t / 3) % 3) * SPAD + t % 3);
    }

    // Stage halo slab (zero padded) into LDS: slab[c][a0][a1][a2], a2 in [0,26)
    for (int idx = threadIdx.x; idx < CC * 3 * 3 * SPAD; idx += 384) {
        int a2  = idx % SPAD;
        int rem = idx / SPAD;
        int a1  = rem % 3; rem /= 3;
        int a0  = rem % 3;
        int c   = rem / 3;
        int z0 = x0 - 1 + a0, z1 = x1 - 1 + a1, z2 = a2 - 1;
        float v = 0.0f;
        if (z0 >= 0 && z0 < DIM && z1 >= 0 && z1 < DIM && z2 >= 0 && z2 < DIM)
            v = xb[c * VOX + z0 * (DIM*DIM) + z1 * DIM + z2];
        slab[idx] = v;
    }
    __syncthreads();

    const int wave = threadIdx.x >> 5;   // 0..11
    const int lane = threadIdx.x & 31;
    const int mt   = wave >> 1;          // 0..5 (rows 0..95, valid o < 81)
    const int nt   = wave & 1;           // 0..1 (cols 0..31, valid x2 < 24)
    const int half = lane >> 4;          // K pair selector per ISA layout
    const int l16  = lane & 15;
    const int x2   = nt * 16 + l16;      // output column (voxel within row)
    const int x2c  = imin(x2, DIM - 1);  // keeps garbage lanes in-bounds
    const int o    = mt * 16 + l16;      // output row (offset channel)

    // Rows >= 81 only feed discarded D rows; clamp pointer for memory safety.
    const v2f* pwo = (const v2f*)(pw + (size_t)imin(o, OC1 - 1) * KDIM);

    v8f acc = {};
#pragma unroll 4
    for (int kb = 0; kb < KDIM / 4; ++kb) {
        const int r0 = 4 * kb + 2 * half;        // K rows (r0, r0+1) for lane
        const unsigned t0 = tab[r0];
        const unsigned t1 = tab[r0 + 1];
        v2f bfrag;
        bfrag.x = slab[t0 + x2c];
        bfrag.y = slab[t1 + x2c];
        v2f afrag = pwo[2 * kb + half];          // contiguous pair, b64 load
        acc = __builtin_amdgcn_wmma_f32_16x16x4_f32(
                  false, afrag, false, bfrag, (short)0, acc, false, false);
    }

    if (x2 < DIM) {
        const int vvox = x0 * (DIM*DIM) + x1 * DIM + x2;
#pragma unroll
        for (int r = 0; r < 8; ++r) {
            int oo = mt * 16 + r + 8 * half;     // C layout: M = r + 8*(lane>=16)
            if (oo < OC1)
                offs[((size_t)b * OC1 + oo) * VOX + vvox] = acc[r] + pb[oo];
        }
    }
}

// ---------------------------------------------------------------------------
// Kernel 2: positions, masks, clips. Writes p_out (d_out region 2) and the
// per-(b,v,n) sample struct {pm0,pm1,pm2, pack(qs0..ql2)} to workspace.
// One thread per (b, voxel). grid = 108 x 256 threads = 27648.
// ---------------------------------------------------------------------------
__global__ __launch_bounds__(256)
void k2_positions(const float* __restrict__ offs, float* __restrict__ pout,
                  float* __restrict__ samp)
{
    const int gid = blockIdx.x * 256 + threadIdx.x;
    if (gid >= BB * VOX) return;
    const int b  = gid / VOX;
    const int vv = gid % VOX;
    const int z[3] = { vv / (DIM*DIM), (vv / DIM) % DIM, vv % DIM };

    const float* offb  = offs + (size_t)b * OC1 * VOX + vv;
    float*       poutv = pout + (size_t)gid * OC1;
    float*       sampv = samp + (size_t)gid * NPTS * 4;

    for (int n = 0; n < NPTS; ++n) {
        const int dig[3] = { n / 9, (n / 3) % 3, n % 3 };
        float pmv[3];
        int   qsv[3], qlv[3];
#pragma unroll
        for (int ax = 0; ax < 3; ++ax) {
            float off = offb[(size_t)(ax * NPTS + n) * VOX];
            // p = (z+1) + (dig-1) + off
            float p  = (float)(z[ax] + dig[ax]) + off;
            float f  = floorf(p);
            int   fi = (int)f;
            bool  m  = (p < 1.0f) || (p > (float)(SPAD - 2));   // < PAD or > s-1-PAD
            float pa = m ? f : p;
            float pm = fminf(fmaxf(pa, 0.0f), (float)(SPAD - 1));
            pmv[ax] = pm;
            qsv[ax] = iclamp(fi,     0, SPAD - 1);
            qlv[ax] = iclamp(fi + 1, 0, SPAD - 1);
            poutv[ax * NPTS + n] = pm;                           // p_out
        }
        sampv[n * 4 + 0] = pmv[0];
        sampv[n * 4 + 1] = pmv[1];
        sampv[n * 4 + 2] = pmv[2];
        unsigned pack = (unsigned)qsv[0] | ((unsigned)qsv[1] << 5) | ((unsigned)qsv[2] << 10)
                      | ((unsigned)qlv[0] << 15) | ((unsigned)qlv[1] << 20) | ((unsigned)qlv[2] << 25);
        ((unsigned*)sampv)[n * 4 + 3] = pack;
    }
}

// ---------------------------------------------------------------------------
// Kernel 3: trilinear gather into LDS x_off tiles + final GEMM
// out[b][o][v] = conv_w[64 x 864] * x_off[864 x v] with fp32 WMMA.
// grid = (576, 2), block = 256 (8 waves: mt = w/2 in [0,4), nt = w&1).
// LDS is built in two 16-channel phases (432 rows x 24 cols = 41.5 KB).
// x_off uses a row-pair-interleaved layout [r/2][col][r&1] so B fragments are
// a single aligned ds_load_b64; A fragments are one b64 load from conv_w.
// ---------------------------------------------------------------------------
__global__ __launch_bounds__(256)
void k3_gather_gemm(const float* __restrict__ x, const float* __restrict__ cw,
                    const float* __restrict__ samp, float* __restrict__ out)
{
    __shared__ float xoff[(CC / 2) * NPTS * DIM];   // 432*24 floats = 41,472 B

    const int b       = blockIdx.y;
    const int rowbase = blockIdx.x * DIM;           // = x0*576 + x1*24
    const float* xb   = x + (size_t)b * CC * VOX;

    const int wave = threadIdx.x >> 5;   // 0..7
    const int lane = threadIdx.x & 31;
    const int mt   = wave >> 1;          // 0..3 -> rows 0..63 (all valid)
    const int nt   = wave & 1;
    const int half = lane >> 4;
    const int l16  = lane & 15;
    const int x2   = nt * 16 + l16;                 // valid if < 24
    const int x2c  = imin(x2, DIM - 1);             // clamped for LDS reads
    const int o    = mt * 16 + l16;

    const v2f* cwo = (const v2f*)(cw + (size_t)o * KDIM);   // o < 64 always valid

    v8f acc = {};
    for (int cphase = 0; cphase < 2; ++cphase) {
        const int cbase = cphase * (CC / 2);

        // ---- phase A: build x_off rows for channels [cbase, cbase+16) ----
        for (int item = threadIdx.x; item < DIM * NPTS; item += 256) {
            const int vloc = item / NPTS;
            const int n    = item % NPTS;
            const float* sv = samp + ((size_t)(b * VOX + rowbase + vloc) * NPTS + n) * 4;
            const float pm0 = sv[0], pm1 = sv[1], pm2 = sv[2];
            const unsigned pack = ((const unsigned*)sv)[3];
            const int q0c[2] = { (int)( pack        & 31), (int)((pack >> 15) & 31) };
            const int q1c[2] = { (int)((pack >>  5) & 31), (int)((pack >> 20) & 31) };
            const int q2c[2] = { (int)((pack >> 10) & 31), (int)((pack >> 25) & 31) };
            const float wt0[2] = { 1.0f + ((float)q0c[0] - pm0), 1.0f - ((float)q0c[1] - pm0) };
            const float wt1[2] = { 1.0f + ((float)q1c[0] - pm1), 1.0f - ((float)q1c[1] - pm1) };
            const float wt2[2] = { 1.0f + ((float)q2c[0] - pm2), 1.0f - ((float)q2c[1] - pm2) };

            float wgt[8];
            int   addr[8];
#pragma unroll
            for (int cr = 0; cr < 8; ++cr) {
                const int b0 = (cr >> 2) & 1, b1 = (cr >> 1) & 1, b2 = cr & 1;
                const int q0 = q0c[b0], q1 = q1c[b1], q2 = q2c[b2];
                // padded border (q==0 or q==25) samples zero -> kill weight
                const bool ok = (q0 >= 1) && (q0 <= DIM) && (q1 >= 1) && (q1 <= DIM)
                             && (q2 >= 1) && (q2 <= DIM);
                wgt[cr]  = ok ? (wt0[b0] * wt1[b1] * wt2[b2]) : 0.0f;
                addr[cr] = ok ? ((q0 - 1) * (DIM*DIM) + (q1 - 1) * DIM + (q2 - 1)) : 0;
            }
            for (int c = 0; c < CC / 2; ++c) {
                const float* xc = xb + (size_t)(cbase + c) * VOX;
                float a = 0.0f;
#pragma unroll
                for (int cr = 0; cr < 8; ++cr)
                    a += wgt[cr] * xc[addr[cr]];
                const int r = c * NPTS + n;          // local K row
                // pair-interleaved: [r/2][vloc][r&1]
                xoff[(r >> 1) * (2 * DIM) + vloc * 2 + (r & 1)] = a;
            }
        }
        __syncthreads();

        // ---- phase B: GEMM over this half of K (432 rows = 108 k-blocks) ----
        const int kbase = cbase * NPTS;             // global K offset (even)
#pragma unroll 4
        for (int kbl = 0; kbl < (CC / 2) * NPTS / 4; ++kbl) {
            const int rl0 = 4 * kbl + 2 * half;     // local row pair base
            v2f bfrag = ((const v2f*)xoff)[(rl0 >> 1) * DIM + x2c];   // b64
            v2f afrag = cwo[(kbase + rl0) >> 1];                      // b64
            acc = __builtin_amdgcn_wmma_f32_16x16x4_f32(
                      false, afrag, false, bfrag, (short)0, acc, false, false);
        }
        __syncthreads();   // WAR guard before xoff is rebuilt
    }

    if (x2 < DIM) {
        const int vvox = rowbase + x2;
#pragma unroll
        for (int r = 0; r < 8; ++r) {
            const int oo = mt * 16 + r + 8 * half;
            out[((size_t)b * OUTC + oo) * VOX + vvox] = acc[r];
        }
    }
}

// ---------------------------------------------------------------------------
// Launch. Inputs: x, p_w, p_b, conv_w (all fp32).
// d_out = out[2,64,24^3] (1,769,472 floats) ++ p_out[2,24^3,81] (2,239,488).
// Workspace: offs (8,957,952 B) then samp structs (11,943,936 B) ~= 20.9 MB.
// ---------------------------------------------------------------------------
extern "C" void kernel_launch(void* const* d_in, const int* in_sizes, int n_in,
                              void* d_out, int out_size, void* d_ws, size_t ws_size,
                              hipStream_t stream) {
    const float* x  = (const float*)d_in[0];
    const float* pw = (const float*)d_in[1];
    const float* pb = (const float*)d_in[2];
    const float* cw = (const float*)d_in[3];

    float* out  = (float*)d_out;
    float* pout = out + (size_t)BB * OUTC * VOX;

    float* offs = (float*)d_ws;
    float* samp = (float*)((char*)d_ws + (size_t)BB * OC1 * VOX * sizeof(float));

    k1_offset_conv<<<dim3(DIM * DIM, BB), 384, 0, stream>>>(x, pw, pb, offs);
    k2_positions <<<dim3((BB * VOX) / 256), 256, 0, stream>>>(offs, pout, samp);
    k3_gather_gemm<<<dim3(DIM * DIM, BB), 256, 0, stream>>>(x, cw, samp, out);
}